// SWTLoss_64372969832981
// MI455X (gfx1250) — compile-verified
//
#include <hip/hip_runtime.h>

// loss = mean((SWT1(x) - SWT1(target))^2)  ==  sum((x-t)^2) / N   (exact identity)
//
// Kernel 1: bandwidth-bound streaming reduction (float4 loads), per-wave
//           reduction via V_WMMA_F32_16X16X4_F32 (A = partials 16x4, B = ones).
// Kernel 2: deterministic fixed-order tree reduction of block partials.

typedef __attribute__((ext_vector_type(2))) float v2f;
typedef __attribute__((ext_vector_type(8))) float v8f;

__global__ __launch_bounds__(256) void swt_mse_partial_kernel(
    const float* __restrict__ x, const float* __restrict__ t,
    float* __restrict__ partials, long long n4, long long n) {
  const long long tid    = (long long)blockIdx.x * 256ll + threadIdx.x;
  const long long stride = (long long)gridDim.x * 256ll;
  const float4* __restrict__ x4 = (const float4*)x;
  const float4* __restrict__ t4 = (const float4*)t;

  float a0 = 0.f, a1 = 0.f, a2 = 0.f, a3 = 0.f;
  for (long long i = tid; i < n4; i += stride) {
    float4 xv = x4[i];
    float4 tv = t4[i];
    float d0 = xv.x - tv.x;
    float d1 = xv.y - tv.y;
    float d2 = xv.z - tv.z;
    float d3 = xv.w - tv.w;
    a0 = fmaf(d0, d0, a0);
    a1 = fmaf(d1, d1, a1);
    a2 = fmaf(d2, d2, a2);
    a3 = fmaf(d3, d3, a3);
  }

  // Scalar tail (n % 4 != 0), handled by block 0 only; reconverges before WMMA.
  if (blockIdx.x == 0) {
    long long base = n4 * 4ll + threadIdx.x;
    if (base < n) {
      float d = x[base] - t[base];
      a0 = fmaf(d, d, a0);
    }
  }

  // ---- Wave32 reduction via matrix engine ----
  // A (16x4 f32): lane L<16 holds A[L][0..1] in {a.x,a.y}; lane L+16 holds A[L][2..3].
  // B = ones(4x16)  =>  D[m][n] = sum_k A[m][k]  (identical for every column n).
  // D layout: lane n<16 has D[0..7][n] in c[0..7]; lane n+16 has D[8..15][n].
  v2f a;
  a.x = a0 + a2;
  a.y = a1 + a3;
  v2f b;
  b.x = 1.0f;
  b.y = 1.0f;
  v8f c = {};
  c = __builtin_amdgcn_wmma_f32_16x16x4_f32(
      /*neg_a=*/false, a, /*neg_b=*/false, b,
      /*c_mod=*/(short)0, c, /*reuse_a=*/false, /*reuse_b=*/false);

  float s8 = c[0] + c[1] + c[2] + c[3] + c[4] + c[5] + c[6] + c[7];
  // lanes n and n+16 together hold the full column sum == wave total
  float wave_total = s8 + __shfl_xor(s8, 16, 32);

  __shared__ float smem[8];
  const int lane = threadIdx.x & 31;
  const int wid  = threadIdx.x >> 5;
  if (lane == 0) smem[wid] = wave_total;
  __syncthreads();
  if (threadIdx.x == 0) {
    float blk = 0.f;
#pragma unroll
    for (int i = 0; i < 8; ++i) blk += smem[i];
    partials[blockIdx.x] = blk;
  }
}

__global__ __launch_bounds__(256) void swt_mse_final_kernel(
    const float* __restrict__ partials, int n_partials,
    float* __restrict__ out, float inv_n) {
  __shared__ float smem[256];
  float s = 0.f;
  for (int i = threadIdx.x; i < n_partials; i += 256) s += partials[i];
  smem[threadIdx.x] = s;
  __syncthreads();
#pragma unroll
  for (int off = 128; off > 0; off >>= 1) {
    if ((int)threadIdx.x < off) smem[threadIdx.x] += smem[threadIdx.x + off];
    __syncthreads();
  }
  if (threadIdx.x == 0) out[0] = smem[0] * inv_n;
}

extern "C" void kernel_launch(void* const* d_in, const int* in_sizes, int n_in,
                              void* d_out, int out_size, void* d_ws, size_t ws_size,
                              hipStream_t stream) {
  const float* x = (const float*)d_in[0];
  const float* t = (const float*)d_in[1];
  float* out = (float*)d_out;
  float* partials = (float*)d_ws;

  const long long n  = (long long)in_sizes[0];  // 16*3*512*512 = 12,582,912
  const long long n4 = n / 4;

  int grid = 2048;  // 2048 blocks * 8 waves: saturates the device; ~6 float4/lane
  // Respect workspace capacity (one float per block partial).
  size_t max_partials = ws_size / sizeof(float);
  if ((size_t)grid > max_partials && max_partials > 0) grid = (int)max_partials;
  if (grid < 1) grid = 1;
  // Don't launch more blocks than there is vector work for.
  long long work_blocks = (n4 + 255) / 256;
  if (work_blocks < 1) work_blocks = 1;
  if ((long long)grid > work_blocks) grid = (int)work_blocks;

  swt_mse_partial_kernel<<<grid, 256, 0, stream>>>(x, t, partials, n4, n);
  swt_mse_final_kernel<<<1, 256, 0, stream>>>(partials, grid, out,
                                              1.0f / (float)n);
}